// DynamicPooling_56006373540031
// MI455X (gfx1250) — compile-verified
//
#include <hip/hip_runtime.h>
#include <math.h>

typedef __attribute__((ext_vector_type(2))) float v2f;
typedef __attribute__((ext_vector_type(4))) float v4f;
typedef __attribute__((ext_vector_type(8))) float v8f;

#define N_ROUT 4096
#define D_DIM  128
#define NT     512          // threads per block (16 wave32s)

__global__ __launch_bounds__(NT) void dyn_routing_kernel(
    const float* __restrict__ E, const int* __restrict__ Tptr,
    float* __restrict__ out)
{
  __shared__ __align__(16) float s_p[N_ROUT];        // logits -> softmax weights
  __shared__ float s_red[NT];                        // block reduction scratch
  __shared__ __align__(16) float s_part[16 * D_DIM]; // phase-3 per-rowslot partials
  __shared__ __align__(16) float s_sig[D_DIM];       // Sigma
  __shared__ __align__(16) float s_w[D_DIM];         // accumulated values (logits = E·w)
  __shared__ float s_scale[1];                       // squash scale

  const int b    = blockIdx.x;
  const int tid  = threadIdx.x;
  const int lane = tid & 31;
  const int lo   = lane & 15;   // M index within 16-row group
  const int hi   = lane >> 4;   // selects K pair {0,1} vs {2,3}
  const int wave = tid >> 5;    // 0..15
  const float* __restrict__ Eb = E + (size_t)b * N_ROUT * D_DIM;
  const int T = *Tptr;

  if (tid < D_DIM) s_w[tid] = 0.0f;
  __syncthreads();

  float v_d = 0.0f; // this thread's component of the last `values` (tid < 128)

  for (int t = 0; t < T; ++t) {
    // ---------- Phase 1: logits[n] = dot(E[n,:], w) via WMMA f32 16x16x4 ----------
    // wave handles rows [wave*256, wave*256+256) in 16 groups of 16 rows.
    for (int g = 0; g < 16; ++g) {
      const int n0 = wave * 256 + g * 16;
      const float* rowp = Eb + (size_t)(n0 + lo) * D_DIM + 2 * hi;
      v8f c0 = {};
      v8f c1 = {};
      #pragma unroll
      for (int q = 0; q < 32; q += 2) {
        // A fragment: lane(lo) = row n0+lo, cols {4q+2hi, 4q+2hi+1}  (b64 load)
        v2f a0 = *(const v2f*)(rowp + 4 * q);
        // B fragment: w broadcast, B[k,j] = w[4q+k] for all j
        v2f b0 = *(const v2f*)(&s_w[4 * q + 2 * hi]);
        c0 = __builtin_amdgcn_wmma_f32_16x16x4_f32(
                 false, a0, false, b0, (short)0, c0, false, false);
        v2f a1 = *(const v2f*)(rowp + 4 * (q + 1));
        v2f b1 = *(const v2f*)(&s_w[4 * (q + 1) + 2 * hi]);
        c1 = __builtin_amdgcn_wmma_f32_16x16x4_f32(
                 false, a1, false, b1, (short)0, c1, false, false);
      }
      // D layout: VGPR r -> M=r (lanes 0-15), M=r+8 (lanes 16-31); every column equal.
      #pragma unroll
      for (int r = 0; r < 8; ++r) {
        float lv = c0[r] + c1[r];
        if (lo == r) s_p[n0 + r + 8 * hi] = lv;
      }
    }
    __syncthreads();

    // ---------- Phase 2: softmax over s_p (block-wide, LDS only) ----------
    float lm = -3.0e38f;
    #pragma unroll
    for (int k = 0; k < N_ROUT / NT; ++k)
      lm = fmaxf(lm, s_p[tid + NT * k]);          // stride-NT: conflict-free
    s_red[tid] = lm;
    __syncthreads();
    for (int s = NT / 2; s > 0; s >>= 1) {
      if (tid < s) s_red[tid] = fmaxf(s_red[tid], s_red[tid + s]);
      __syncthreads();
    }
    const float m = s_red[0];
    __syncthreads();

    float ls = 0.0f;
    #pragma unroll
    for (int k = 0; k < N_ROUT / NT; ++k) {
      float p = __expf(s_p[tid + NT * k] - m);
      s_p[tid + NT * k] = p;
      ls += p;
    }
    s_red[tid] = ls;
    __syncthreads();
    for (int s = NT / 2; s > 0; s >>= 1) {
      if (tid < s) s_red[tid] += s_red[tid + s];
      __syncthreads();
    }
    const float inv_s = 1.0f / s_red[0];
    __syncthreads();

    // ---------- Phase 3: Sigma[d] = (1/s) * sum_n p[n] * E[n,d] ----------
    // thread owns 4 columns (b128 loads); 16 row-slots interleave rows:
    // per step the block streams 16 consecutive rows = 8 KB contiguous.
    {
      const int c4 = (tid & 31) * 4;  // column base
      const int rs = tid >> 5;        // row slot 0..15 (== wave)
      v4f acc = {};
      #pragma unroll 4
      for (int n = rs; n < N_ROUT; n += 16) {
        v4f e = *(const v4f*)(Eb + (size_t)n * D_DIM + c4);
        float p = s_p[n];
        acc.x = fmaf(p, e.x, acc.x);
        acc.y = fmaf(p, e.y, acc.y);
        acc.z = fmaf(p, e.z, acc.z);
        acc.w = fmaf(p, e.w, acc.w);
      }
      *(v4f*)(&s_part[rs * D_DIM + c4]) = acc;
    }
    __syncthreads();
    if (tid < D_DIM) {
      float sum = 0.0f;
      #pragma unroll
      for (int r = 0; r < 16; ++r) sum += s_part[r * D_DIM + tid];
      s_sig[tid] = sum * inv_s;
    }
    __syncthreads();

    // ---------- Squash: v = Sigma * ||Sigma|| / (1 + ||Sigma||^2) ----------
    if (wave == 0) {
      float sq = 0.0f;
      #pragma unroll
      for (int k = 0; k < 4; ++k) {
        float x = s_sig[lane * 4 + k];
        sq += x * x;
      }
      #pragma unroll
      for (int off = 16; off > 0; off >>= 1)
        sq += __shfl_xor(sq, off, 32);
      if (lane == 0) {
        float ns = sq;
        s_scale[0] = sqrtf(ns) / (1.0f + ns);
      }
    }
    __syncthreads();
    if (tid < D_DIM) {
      float vv = s_sig[tid] * s_scale[0];
      v_d = vv;
      s_w[tid] += vv;   // telescoped B_logits: logits_next = E · w
    }
    __syncthreads();
  }

  if (tid < D_DIM) out[(size_t)b * D_DIM + tid] = v_d;
}

extern "C" void kernel_launch(void* const* d_in, const int* in_sizes, int n_in,
                              void* d_out, int out_size, void* d_ws, size_t ws_size,
                              hipStream_t stream) {
  const float* E = (const float*)d_in[0];
  const int*   T = (const int*)d_in[1];
  float* out = (float*)d_out;
  const int B = in_sizes[0] / (N_ROUT * D_DIM);
  dyn_routing_kernel<<<B, NT, 0, stream>>>(E, T, out);
}